// ChunkedEmbedding_27255862460962
// MI455X (gfx1250) — compile-verified
//
#include <hip/hip_runtime.h>

// Embedding gather for MI455X (gfx1250, wave32).
//
// out[row, :] = weight[indices[row], :],  row in [0, 4096*200), EMBED_DIM=128.
//
// Bandwidth-bound: ~419 MB stored, gather reads serviced from L2 (table is
// 51.2 MB < 192 MB L2). Output stores are non-temporal so the streamed output
// does not evict the L2-resident embedding table.
//
// Layout: one wave per 4 rows; each row is 512 B = 32 lanes x 16 B, so each
// lane does exactly one global_load_b128 + one global_store_b128(NT) per row.
// The 4 indices per wave are wave-uniform -> single s_load_b128 on KMcnt.

#define NUM_ROWS   (4096 * 200)   // BATCH * HIST = 819200
#define EMBED_DIM  128
#define WAVE_SIZE  32
#define WAVES_PER_BLOCK 8
#define ROWS_PER_WAVE   4
#define ROWS_PER_BLOCK  (WAVES_PER_BLOCK * ROWS_PER_WAVE)   // 32

typedef __attribute__((ext_vector_type(4))) float v4f;

__global__ __launch_bounds__(WAVES_PER_BLOCK * WAVE_SIZE)
void chunked_embedding_gather(const unsigned int* __restrict__ indices,
                              const float* __restrict__ weight,
                              float* __restrict__ out) {
    // Wave-uniform wave id: readfirstlane pins it to an SGPR so the row ids
    // and index addresses are provably uniform -> scalar s_load for indices.
    const unsigned int wave =
        (unsigned int)__builtin_amdgcn_readfirstlane((int)(threadIdx.x >> 5));
    const unsigned int lane = threadIdx.x & (WAVE_SIZE - 1);

    // Grid covers exactly NUM_ROWS rows (25600 blocks x 32 rows); no guard.
    const size_t base_row =
        ((size_t)blockIdx.x * WAVES_PER_BLOCK + (size_t)wave) * ROWS_PER_WAVE;

    // 4 wave-uniform, consecutive index fetches -> one s_load_b128.
    // Unsigned: zero-extend, no s_ashr sign-extension pairs.
    unsigned int e[ROWS_PER_WAVE];
#pragma unroll
    for (int r = 0; r < ROWS_PER_WAVE; ++r) {
        e[r] = indices[base_row + (size_t)r];
    }

    // 4 independent 16B gather loads per lane (32 lanes x 16B = one 512B row
    // per wave). Default temporal policy keeps the table resident in L2.
    v4f v[ROWS_PER_WAVE];
#pragma unroll
    for (int r = 0; r < ROWS_PER_WAVE; ++r) {
        const v4f* __restrict__ src =
            (const v4f*)(weight + (size_t)e[r] * EMBED_DIM);
        v[r] = src[lane];
    }

    // Non-temporal streamed stores: the 419 MB output must not evict the
    // 51 MB embedding table from the 192 MB L2.
#pragma unroll
    for (int r = 0; r < ROWS_PER_WAVE; ++r) {
        v4f* dst = (v4f*)(out + (base_row + (size_t)r) * EMBED_DIM);
        __builtin_nontemporal_store(v[r], dst + lane);
    }
}

extern "C" void kernel_launch(void* const* d_in, const int* in_sizes, int n_in,
                              void* d_out, int out_size, void* d_ws, size_t ws_size,
                              hipStream_t stream) {
    (void)in_sizes; (void)n_in; (void)out_size; (void)d_ws; (void)ws_size;

    const unsigned int* indices = (const unsigned int*)d_in[0]; // [4096,200] int32 (non-negative)
    const float*        weight  = (const float*)d_in[1];        // [100000,128] fp32
    float*              out     = (float*)d_out;                // [4096,200,128] fp32

    const int grid = NUM_ROWS / ROWS_PER_BLOCK;   // 819200 / 32 = 25600
    chunked_embedding_gather<<<grid, WAVES_PER_BLOCK * WAVE_SIZE, 0, stream>>>(
        indices, weight, out);
}